// GridSamplePScan_29085518528765
// MI455X (gfx1250) — compile-verified
//
#include <hip/hip_runtime.h>
#include <math.h>

// ---------------------------------------------------------------------------
// GridSamplePScan for MI455X (gfx1250, wave32)
//   B=1, L=64, C=16, H=W=64
//   out[t] = sum_{k<=t} exp(-decay*(t-k)) * bilinear(images[k], base + cum[t]-cum[k])
//
// Kernel 1: cum = tril(ones) @ flows via V_WMMA_F32_16X16X4_F32 (exact f32),
//           stored interleaved as [t, pixel, {x,y}] for one-b64 loads later.
// Kernel 2: images [L,C,H,W] -> [L,H,W,C] so a bilinear corner = 64B contiguous
// Kernel 3: gather/accumulate sampler, 1 thread = (t, pixel), 16 ch in VGPRs
// ---------------------------------------------------------------------------

typedef float v2f __attribute__((ext_vector_type(2)));
typedef float v8f __attribute__((ext_vector_type(8)));

#define L_DIM   64
#define C_DIM   16
#define HW      4096            // 64*64
#define P_DIM   8192            // 2*HW (x,y flow planes per timestep)

// ---------------------------------------------------------------------------
// Kernel 1: cumulative sum over time as a lower-triangular f32 GEMM.
//   cum[t, col] = sum_k tril[t,k] * flows[k, col],  t,k in [0,64), col in [0,8192)
// One wave computes a 16(t) x 16(col) tile; K advances in steps of 4.
// VGPR layouts per CDNA5 ISA 7.12.2:
//   A 16x4 : lanes 0-15 M=lane, VGPR0 holds K=(0|2) by lane-half, VGPR1 K=(1|3)
//   B 4x16 : lanes 0-15 N=lane, VGPR0 holds K=(0|2) by lane-half, VGPR1 K=(1|3)
//   C/D    : VGPR r -> row r (lanes 0-15) / row r+8 (lanes 16-31), N=lane&15
// Output is scattered into interleaved layout cum_xy[t][pix][{x,y}]:
//   col < 4096 -> x plane, col >= 4096 -> y plane (a 16-col tile never straddles)
// ---------------------------------------------------------------------------
__global__ void cumsum_wmma_kernel(const float* __restrict__ flows,
                                   float* __restrict__ cum_xy) {
  const int wave  = (blockIdx.x * blockDim.x + threadIdx.x) >> 5;
  const int lane  = threadIdx.x & 31;
  const int tTile = wave & 3;          // 4 tiles over t (64/16)
  const int pTile = wave >> 2;         // 512 tiles over columns (8192/16)
  const int t0    = tTile << 4;
  const int p0    = pTile << 4;
  const int half  = lane >> 4;         // 0: lanes 0-15, 1: lanes 16-31
  const int mn    = lane & 15;         // M index for A, N index for B/D

  v8f c = {};
#if __has_builtin(__builtin_amdgcn_wmma_f32_16x16x4_f32)
  for (int k0 = 0; k0 <= t0 + 15; k0 += 4) {   // tiles with k0 > t0+15 are all-zero
    const int kA = k0 + (half << 1);
    v2f a, b;
    a.x = (kA     <= t0 + mn) ? 1.0f : 0.0f;   // tril mask
    a.y = (kA + 1 <= t0 + mn) ? 1.0f : 0.0f;
    b.x = flows[(size_t)kA       * P_DIM + p0 + mn];
    b.y = flows[(size_t)(kA + 1) * P_DIM + p0 + mn];
    c = __builtin_amdgcn_wmma_f32_16x16x4_f32(
        /*neg_a=*/false, a, /*neg_b=*/false, b,
        /*c_mod=*/(short)0, c, /*reuse_a=*/false, /*reuse_b=*/false);
  }
#else
  // Scalar fallback (should not be taken on gfx1250)
  for (int k = 0; k <= t0 + 15; ++k) {
    float v = flows[(size_t)k * P_DIM + p0 + mn];
    #pragma unroll
    for (int r = 0; r < 8; ++r) {
      int t = t0 + r + (half << 3);
      if (k <= t) c[r] += v;
    }
  }
#endif

  const int col   = p0 + mn;
  const int plane = col >> 12;         // 0 = x flow, 1 = y flow
  const int pix   = col & (HW - 1);
  #pragma unroll
  for (int r = 0; r < 8; ++r) {
    const int t = t0 + r + (half << 3);
    cum_xy[(((size_t)t * HW + pix) << 1) + plane] = c[r];
  }
}

// ---------------------------------------------------------------------------
// Kernel 2: images [64,16,64,64] -> imgT [64,64,64,16]  (channel-innermost)
// Writes are fully coalesced; reads strided (one-shot, ~17MB, negligible).
// ---------------------------------------------------------------------------
__global__ void transpose_hwc_kernel(const float* __restrict__ img,
                                     float* __restrict__ imgT) {
  const int idx = blockIdx.x * 256 + threadIdx.x;    // over 64*16*4096
  const int c   = idx & 15;
  const int yx  = (idx >> 4) & (HW - 1);
  const int k   = idx >> 16;                         // /(4096*16)
  imgT[idx] = img[((size_t)(k * C_DIM + c)) * HW + yx];
}

// ---------------------------------------------------------------------------
// Kernel 3: weighted bilinear scan.
// ---------------------------------------------------------------------------
__device__ __forceinline__ void gather_fma16(const float4* __restrict__ p,
                                             float w, float acc[16]) {
  #pragma unroll
  for (int i = 0; i < 4; ++i) {
    float4 v = p[i];                       // global_load_b128, 64B-aligned
    acc[4 * i + 0] = fmaf(w, v.x, acc[4 * i + 0]);
    acc[4 * i + 1] = fmaf(w, v.y, acc[4 * i + 1]);
    acc[4 * i + 2] = fmaf(w, v.z, acc[4 * i + 2]);
    acc[4 * i + 3] = fmaf(w, v.w, acc[4 * i + 3]);
  }
}

__global__ void sampler_kernel(const float2* __restrict__ cum_xy,
                               const float* __restrict__ imgT,
                               const float* __restrict__ decay_log,
                               float* __restrict__ out) {
  const int t = blockIdx.x >> 4;                         // 16 blocks of 256 per t
  const int p = ((blockIdx.x & 15) << 8) + threadIdx.x;  // pixel 0..4095
  const int wx = p & 63;
  const int hy = p >> 6;

  const float bx = (float)(2 * wx + 1) * (1.0f / 64.0f) - 1.0f;  // base grid x
  const float by = (float)(2 * hy + 1) * (1.0f / 64.0f) - 1.0f;  // base grid y

  const float2 ct = cum_xy[(size_t)t * HW + p];          // cum[t] (x,y) at p

  const float decay = expf(decay_log[0]);
  const float stepw = expf(-decay);

  float acc[16];
  #pragma unroll
  for (int i = 0; i < 16; ++i) acc[i] = 0.0f;

  float wgt = 1.0f;                                      // exp(-decay*(t-k)), k=t..0
  for (int k = t; k >= 0; --k) {
    const float2 ck = cum_xy[(size_t)k * HW + p];        // one global_load_b64
    const float relx = ct.x - ck.x;
    const float rely = ct.y - ck.y;

    // wrap x periodically into [-1, 1): remainder(g+1, 2) - 1
    float gxn = bx + relx + 1.0f;
    gxn = gxn - floorf(gxn * 0.5f) * 2.0f;               // [0, 2)
    const float gx = gxn * 32.0f - 0.5f;                 // ((gxn-1)+1)*W/2 - 0.5
    const float gy = (by + rely + 1.0f) * 32.0f - 0.5f;

    const float x0f = floorf(gx);
    const float y0f = floorf(gy);
    const float fx = gx - x0f;
    const float fy = gy - y0f;
    const int x0 = (int)x0f;                             // in [-1, 63] after wrap
    const int y0 = (int)y0f;

    const float w00 = wgt * (1.0f - fx) * (1.0f - fy);
    const float w01 = wgt * fx * (1.0f - fy);
    const float w10 = wgt * (1.0f - fx) * fy;
    const float w11 = wgt * fx * fy;

    const float* base = imgT + ((size_t)k << 16);        // k * 4096 * 16
    const bool xv0 = (x0 >= 0) && (x0 < 64);
    const bool xv1 = (x0 + 1 < 64);                      // x0+1 >= 0 always
    const bool yv0 = (y0 >= 0) && (y0 < 64);
    const bool yv1 = (y0 + 1 >= 0) && (y0 + 1 < 64);

    if (yv0) {
      const float* row = base + ((size_t)y0 << 10);      // y0 * 64 * 16
      if (xv0) gather_fma16((const float4*)(row + ((size_t)x0 << 4)), w00, acc);
      if (xv1) gather_fma16((const float4*)(row + ((size_t)(x0 + 1) << 4)), w01, acc);
    }
    if (yv1) {
      const float* row = base + ((size_t)(y0 + 1) << 10);
      if (xv0) gather_fma16((const float4*)(row + ((size_t)x0 << 4)), w10, acc);
      if (xv1) gather_fma16((const float4*)(row + ((size_t)(x0 + 1) << 4)), w11, acc);
    }
    wgt *= stepw;
  }

  // out layout [1, 64, 16, 64, 64]
  #pragma unroll
  for (int c = 0; c < 16; ++c)
    out[((size_t)(t * C_DIM + c)) * HW + p] = acc[c];
}

// ---------------------------------------------------------------------------
// Launch
// ---------------------------------------------------------------------------
extern "C" void kernel_launch(void* const* d_in, const int* in_sizes, int n_in,
                              void* d_out, int out_size, void* d_ws, size_t ws_size,
                              hipStream_t stream) {
  const float* flows     = (const float*)d_in[0];   // [1,64,2,64,64]
  const float* images    = (const float*)d_in[1];   // [1,64,16,64,64]
  const float* decay_log = (const float*)d_in[2];   // scalar
  float* out = (float*)d_out;                       // [1,64,16,64,64]

  float* cum_xy = (float*)d_ws;                     // 64*4096*2 f32 = 2 MB
  float* imgT   = cum_xy + (size_t)L_DIM * P_DIM;   // 64*4096*16 f32 = 16.8 MB

  // 2048 waves: 4 t-tiles x 512 col-tiles; 4 waves per 128-thread block
  cumsum_wmma_kernel<<<512, 128, 0, stream>>>(flows, cum_xy);
  // 64*16*4096 elements / 256
  transpose_hwc_kernel<<<16384, 256, 0, stream>>>(images, imgT);
  // 64 t * 4096 px, 256 threads/block
  sampler_kernel<<<64 * 16, 256, 0, stream>>>((const float2*)cum_xy, imgT,
                                              decay_log, out);
}